// LineVectorizer_34754875359769
// MI455X (gfx1250) — compile-verified
//
#include <hip/hip_runtime.h>

typedef __attribute__((ext_vector_type(2))) float v2f;
typedef __attribute__((ext_vector_type(8))) float v8f;

#define HW    128
#define NPIX  (HW * HW)        // 16384 pixels per image
#define CIN   256
#define DIM   128              // dim_loi
#define PL    64               // planes
#define BATCH 8
#define LPB   2048             // lines per image
#define NLINES (BATCH * LPB)
#define NPTS0 32
#define NPTS1 8
#define EPSV  1e-5f

// ---------------------------------------------------------------------------
// Kernel A: fc1 (1x1 conv 256->128) as fp32 WMMA GEMM (unchanged; codegen good).
//   out[b][pixel][o] = sum_c w[o][c] * feat[b][c][pixel] + bias[o]
// ---------------------------------------------------------------------------
__global__ __launch_bounds__(256)
void fc1_wmma_kernel(const float* __restrict__ feat,
                     const float* __restrict__ w,
                     const float* __restrict__ bias,
                     float* __restrict__ xout) {
    const int b       = blockIdx.y;
    const int colbase = blockIdx.x * 16;
    const int wave    = threadIdx.x >> 5;
    const int lane    = threadIdx.x & 31;
    const int half    = lane >> 4;
    const int mn      = lane & 15;
    const int obase   = wave * 16;

    const float* wp = w + (size_t)(obase + mn) * CIN + 2 * half;
    const float* fp = feat + ((size_t)b * CIN + 2 * half) * NPIX + colbase + mn;

    v8f c = {};
#pragma unroll 4
    for (int kk = 0; kk < CIN; kk += 4) {
        v2f a;
        a.x = wp[kk];
        a.y = wp[kk + 1];
        v2f bv;
        bv.x = fp[(size_t)kk * NPIX];
        bv.y = fp[(size_t)(kk + 1) * NPIX];
        c = __builtin_amdgcn_wmma_f32_16x16x4_f32(false, a, false, bv,
                                                  (short)0, c, false, false);
    }

    const int col  = colbase + mn;
    const int orow = obase + 8 * half;
    float* op = xout + ((size_t)b * NPIX + col) * DIM + orow;
#pragma unroll
    for (int r = 0; r < 8; ++r)
        op[r] = c[r] + bias[orow + r];
}

// ---------------------------------------------------------------------------
// Kernel B: 2 lines per block (N = 16 columns), 128 threads = 4 waves.
// sampling/maxpool -> LDS, then conv1 / conv2(3 shifted GEMMs) / conv3 all as
// fp32 WMMA with LDS-resident B operands. Deterministic reduction for fc2.
// Column index lt = line*8 + t.
// ---------------------------------------------------------------------------
__global__ __launch_bounds__(128)
void loi_head_wmma_kernel(const float* __restrict__ x,
                          const float* __restrict__ lines,
                          const float* __restrict__ bn1g, const float* __restrict__ bn1b,
                          const float* __restrict__ bn1m, const float* __restrict__ bn1v,
                          const float* __restrict__ w1,   const float* __restrict__ b1,
                          const float* __restrict__ bn2g, const float* __restrict__ bn2b,
                          const float* __restrict__ bn2m, const float* __restrict__ bn2v,
                          const float* __restrict__ w2,   const float* __restrict__ b2,
                          const float* __restrict__ bn3g, const float* __restrict__ bn3b,
                          const float* __restrict__ bn3m, const float* __restrict__ bn3v,
                          const float* __restrict__ w3,   const float* __restrict__ b3,
                          const float* __restrict__ fc2w, const float* __restrict__ fc2b,
                          float* __restrict__ out) {
    __shared__ float s_xp [DIM * 16];   // pooled features [c][lt] (residual)
    __shared__ float s_h  [DIM * 16];   // relu(bn1(xp))
    __shared__ float s_h2 [PL * 16];    // after conv1+bn2+relu
    __shared__ float s_h2m[PL * 16];    // s_h2 shifted t-1 (zero pad)
    __shared__ float s_h2p[PL * 16];    // s_h2 shifted t+1 (zero pad)
    __shared__ float s_h3 [PL * 16];    // after conv2+bn3+relu
    __shared__ float s_bn2s[PL], s_bn2o[PL], s_bn3s[PL], s_bn3o[PL];
    __shared__ float s_part[128];

    const int tid  = threadIdx.x;
    const int wave = tid >> 5;
    const int lane = tid & 31;
    const int half = lane >> 4;
    const int mn   = lane & 15;

    // ---- per-channel BN scale/offset tables ----
    if (tid < PL) {
        const float s2 = bn2g[tid] * rsqrtf(bn2v[tid] + EPSV);
        s_bn2s[tid] = s2;
        s_bn2o[tid] = bn2b[tid] - bn2m[tid] * s2;
        const float s3 = bn3g[tid] * rsqrtf(bn3v[tid] + EPSV);
        s_bn3s[tid] = s3;
        s_bn3o[tid] = bn3b[tid] - bn3m[tid] * s3;
    }

    // ---- sample both lines: bilinear + maxpool + bn1/relu (thread = channel) ----
    {
        const int   c  = tid;
        const float s1 = bn1g[c] * rsqrtf(bn1v[c] + EPSV);
        const float o1 = bn1b[c] - bn1m[c] * s1;
        for (int li = 0; li < 2; ++li) {
            const int n = blockIdx.x * 2 + li;
            const int b = n / LPB;
            const float e0x = lines[(size_t)n * 4 + 0];
            const float e0y = lines[(size_t)n * 4 + 1];
            const float e1x = lines[(size_t)n * 4 + 2];
            const float e1y = lines[(size_t)n * 4 + 3];
            const float* xb = x + (size_t)b * NPIX * DIM + c;

            float mx[NPTS1];
#pragma unroll
            for (int t = 0; t < NPTS1; ++t) mx[t] = -3.4e38f;

#pragma unroll 4
            for (int p = 0; p < NPTS0; ++p) {
                const float lam = (float)p * (1.0f / 31.0f);
                const float px  = e0x * lam + e1x * (1.0f - lam) - 0.5f;
                const float py  = e0y * lam + e1y * (1.0f - lam) - 0.5f;
                const float fx0 = fminf(fmaxf(floorf(px), 0.0f), 127.0f);
                const float fy0 = fminf(fmaxf(floorf(py), 0.0f), 127.0f);
                const float fx1 = fminf(fx0 + 1.0f, 127.0f);
                const float fy1 = fminf(fy0 + 1.0f, 127.0f);
                const int ix0 = (int)fx0, iy0 = (int)fy0;
                const int ix1 = (int)fx1, iy1 = (int)fy1;
                const float w00 = (fx1 - px) * (fy1 - py);
                const float w10 = (px - fx0) * (fy1 - py);
                const float w01 = (fx1 - px) * (py - fy0);
                const float w11 = (px - fx0) * (py - fy0);
                const float v = w00 * xb[(size_t)(ix0 * HW + iy0) * DIM]
                              + w10 * xb[(size_t)(ix1 * HW + iy0) * DIM]
                              + w01 * xb[(size_t)(ix0 * HW + iy1) * DIM]
                              + w11 * xb[(size_t)(ix1 * HW + iy1) * DIM];
                mx[p >> 2] = fmaxf(mx[p >> 2], v);
            }
#pragma unroll
            for (int t = 0; t < NPTS1; ++t) {
                const int lt = li * NPTS1 + t;
                s_xp[c * 16 + lt] = mx[t];
                s_h [c * 16 + lt] = fmaxf(mx[t] * s1 + o1, 0.0f);
            }
        }
    }
    __syncthreads();

    // ---- conv1: WMMA GEMM 64x128x16; wave -> M strip [16w, 16w+16) ----
    {
        const float* wp = w1 + (size_t)(wave * 16 + mn) * DIM + 2 * half;
        v8f c = {};
#pragma unroll 4
        for (int kk = 0; kk < DIM; kk += 4) {
            v2f a;
            a.x = wp[kk];
            a.y = wp[kk + 1];
            v2f bv;
            bv.x = s_h[(kk + 2 * half) * 16 + mn];
            bv.y = s_h[(kk + 2 * half + 1) * 16 + mn];
            c = __builtin_amdgcn_wmma_f32_16x16x4_f32(false, a, false, bv,
                                                      (short)0, c, false, false);
        }
        const int obase = wave * 16 + 8 * half;
#pragma unroll
        for (int r = 0; r < 8; ++r) {
            const int o = obase + r;
            s_h2[o * 16 + mn] =
                fmaxf((c[r] + b1[o]) * s_bn2s[o] + s_bn2o[o], 0.0f);
        }
    }
    __syncthreads();

    // ---- build t-1 / t+1 shifted copies with zero padding (per line) ----
    for (int idx = tid; idx < PL * 16; idx += 128) {
        const int t = idx & 7;   // t within the line's 8-column group
        s_h2m[idx] = (t > 0) ? s_h2[idx - 1] : 0.0f;
        s_h2p[idx] = (t < 7) ? s_h2[idx + 1] : 0.0f;
    }
    __syncthreads();

    // ---- conv2: 3 shifted WMMA GEMMs 64x64x16, one accumulator ----
    {
        const size_t wb = (size_t)(wave * 16 + mn) * PL * 3;
        v8f c = {};
#pragma unroll 2
        for (int kk = 0; kk < PL; kk += 4) {
            const int k0 = kk + 2 * half;
            const int k1 = k0 + 1;
            v2f a, bv;
            // tap 0 (t-1)
            a.x = w2[wb + (size_t)k0 * 3 + 0];
            a.y = w2[wb + (size_t)k1 * 3 + 0];
            bv.x = s_h2m[k0 * 16 + mn];
            bv.y = s_h2m[k1 * 16 + mn];
            c = __builtin_amdgcn_wmma_f32_16x16x4_f32(false, a, false, bv,
                                                      (short)0, c, false, false);
            // tap 1 (t)
            a.x = w2[wb + (size_t)k0 * 3 + 1];
            a.y = w2[wb + (size_t)k1 * 3 + 1];
            bv.x = s_h2[k0 * 16 + mn];
            bv.y = s_h2[k1 * 16 + mn];
            c = __builtin_amdgcn_wmma_f32_16x16x4_f32(false, a, false, bv,
                                                      (short)0, c, false, false);
            // tap 2 (t+1)
            a.x = w2[wb + (size_t)k0 * 3 + 2];
            a.y = w2[wb + (size_t)k1 * 3 + 2];
            bv.x = s_h2p[k0 * 16 + mn];
            bv.y = s_h2p[k1 * 16 + mn];
            c = __builtin_amdgcn_wmma_f32_16x16x4_f32(false, a, false, bv,
                                                      (short)0, c, false, false);
        }
        const int obase = wave * 16 + 8 * half;
#pragma unroll
        for (int r = 0; r < 8; ++r) {
            const int o = obase + r;
            s_h3[o * 16 + mn] =
                fmaxf((c[r] + b2[o]) * s_bn3s[o] + s_bn3o[o], 0.0f);
        }
    }
    __syncthreads();

    // ---- conv3: WMMA GEMM 128x64x16 (2 M-tiles per wave) + residual + fc2 ----
    {
        const int t = mn & 7;   // t within line; line = mn >> 3
        float partial = 0.0f;
#pragma unroll
        for (int pass = 0; pass < 2; ++pass) {
            const int mt = wave + pass * 4;
            const float* wp = w3 + (size_t)(mt * 16 + mn) * PL + 2 * half;
            v8f c = {};
#pragma unroll 4
            for (int kk = 0; kk < PL; kk += 4) {
                v2f a;
                a.x = wp[kk];
                a.y = wp[kk + 1];
                v2f bv;
                bv.x = s_h3[(kk + 2 * half) * 16 + mn];
                bv.y = s_h3[(kk + 2 * half + 1) * 16 + mn];
                c = __builtin_amdgcn_wmma_f32_16x16x4_f32(false, a, false, bv,
                                                          (short)0, c, false, false);
            }
            const int obase = mt * 16 + 8 * half;
#pragma unroll
            for (int r = 0; r < 8; ++r) {
                const int o  = obase + r;
                const float xn = s_xp[o * 16 + mn] + c[r] + b3[o];
                partial += fmaxf(xn, 0.0f) * fc2w[o * NPTS1 + t];
            }
        }
        s_part[tid] = partial;   // this lane's contribution to line (mn>>3)
    }
    __syncthreads();

    // ---- deterministic per-line reduction (line = bit3 of tid) ----
    if (tid < 2) {
        float acc = 0.0f;
        for (int j = 0; j < 128; ++j)
            if (((j >> 3) & 1) == tid) acc += s_part[j];
        out[blockIdx.x * 2 + tid] = acc + fc2b[0];
    }
}

// ---------------------------------------------------------------------------
extern "C" void kernel_launch(void* const* d_in, const int* in_sizes, int n_in,
                              void* d_out, int out_size, void* d_ws, size_t ws_size,
                              hipStream_t stream) {
    const float* feature = (const float*)d_in[0];
    const float* lines   = (const float*)d_in[1];
    const float* fc1_w   = (const float*)d_in[2];
    const float* fc1_b   = (const float*)d_in[3];
    const float* bn1g = (const float*)d_in[4],  *bn1b = (const float*)d_in[5];
    const float* bn1m = (const float*)d_in[6],  *bn1v = (const float*)d_in[7];
    const float* w1   = (const float*)d_in[8],  *b1   = (const float*)d_in[9];
    const float* bn2g = (const float*)d_in[10], *bn2b = (const float*)d_in[11];
    const float* bn2m = (const float*)d_in[12], *bn2v = (const float*)d_in[13];
    const float* w2   = (const float*)d_in[14], *b2   = (const float*)d_in[15];
    const float* bn3g = (const float*)d_in[16], *bn3b = (const float*)d_in[17];
    const float* bn3m = (const float*)d_in[18], *bn3v = (const float*)d_in[19];
    const float* w3   = (const float*)d_in[20], *b3   = (const float*)d_in[21];
    const float* fc2w = (const float*)d_in[22], *fc2b = (const float*)d_in[23];

    float* xbuf = (float*)d_ws;   // [B][16384 pixels][128 ch] = 64 MiB

    dim3 g1(NPIX / 16, BATCH);
    fc1_wmma_kernel<<<g1, 256, 0, stream>>>(feature, fc1_w, fc1_b, xbuf);

    loi_head_wmma_kernel<<<NLINES / 2, 128, 0, stream>>>(
        xbuf, lines,
        bn1g, bn1b, bn1m, bn1v, w1, b1,
        bn2g, bn2b, bn2m, bn2v, w2, b2,
        bn3g, bn3b, bn3m, bn3v, w3, b3,
        fc2w, fc2b, (float*)d_out);
}